// SurfaceNormalOptimizer_67894843015588
// MI455X (gfx1250) — compile-verified
//
#include <hip/hip_runtime.h>

typedef __attribute__((ext_vector_type(4))) float v4f;

namespace {
constexpr int kB  = 8;
constexpr int kH  = 384;
constexpr int kW  = 1280;
constexpr int kW4 = kW / 4;              // 320 float4 strips per row
constexpr int kHW = kH * kW;             // plane size
}

// Grid: (1, H, B). Block: 320 threads = 10 wave32 = exactly one image row.
// b and y are SGPR-resident (blockIdx), so intrinsics load via SMEM and all
// row/plane addressing is scalar. One thread = one 16B-aligned 4-pixel strip.
__global__ __launch_bounds__(320) void SurfaceNormalOptimizer_kernel(
    const float* __restrict__ depth,   // (B,1,H,W) f32
    const float* __restrict__ intr,    // (B,3,3)   f32
    float* __restrict__ out)           // (B,3,H,W) f32
{
    const int b  = blockIdx.z;         // uniform -> SGPR
    const int y  = blockIdx.y;         // uniform -> SGPR
    const int x0 = threadIdx.x * 4;    // 0..1276

    const float* dp = depth + ((size_t)b * kH + y) * kW + x0;

    // 128-bit vector loads; the row below is re-served from L2 when the
    // owning block reads it as 'd'.
    v4f   d4   = *(const v4f*)dp;
    float dnxt = (x0 + 4 < kW) ? dp[4] : 0.0f;            // zero-pad right edge
    v4f   dnv  = (y + 1 < kH) ? *(const v4f*)(dp + kW)    // zero-pad bottom edge
                              : (v4f)(0.0f);              // (uniform branch)

    // Uniform address -> scalar (SMEM) loads, broadcast to the wave.
    const float fx = intr[b * 9 + 0];
    const float bx = intr[b * 9 + 2];
    const float fy = intr[b * 9 + 4];
    const float by = intr[b * 9 + 5];
    const float rfx = 1.0f / fx;
    const float rfy = 1.0f / fy;
    const float vv  = ((float)y - by) * rfy;              // uniform

    float dd [4] = {d4.x,  d4.y,  d4.z,  d4.w};
    float drt[4] = {d4.y,  d4.z,  d4.w,  dnxt};   // right neighbors
    float dbl[4] = {dnv.x, dnv.y, dnv.z, dnv.w};  // below neighbors

    constexpr float kLim = 1.0f - 1e-6f;
    float r0[4], r1[4], r2[4];

#pragma unroll
    for (int k = 0; k < 4; ++k) {
        float d  = dd[k];
        float gx = drt[k] - d;               // gradx (0-pad => -d at edge)
        float gy = dbl[k] - d;               // grady
        float u  = ((float)(x0 + k) - bx) * rfx;

        float vx1 = d * rfx + u * gx;        // vx = (vx1, vx2, gx)
        float vx2 = vv * gx;
        float vy1 = u * gy;                  // vy = (vy1, vy2, gy)
        float vy2 = d * rfy + vv * gy;

        float n0 = vx2 * gy  - gx  * vy2;    // cross(vx, vy)
        float n1 = gx  * vy1 - vx1 * gy;
        float n2 = vx1 * vy2 - vx2 * vy1;

        float s   = n0 * n0 + n1 * n1 + n2 * n2;
        // rsq(max(s,1e-24)) == 1/max(sqrt(s),1e-12); single v_rsq_f32.
        float inv = __builtin_amdgcn_rsqf(fmaxf(s, 1e-24f));

        r0[k] = fminf(fmaxf(n0 * inv, -kLim), kLim);
        r1[k] = fminf(fmaxf(n1 * inv, -kLim), kLim);
        r2[k] = fminf(fmaxf(n2 * inv, -kLim), kLim);
    }

    v4f o0 = (v4f){r0[0], r0[1], r0[2], r0[3]};
    v4f o1 = (v4f){r1[0], r1[1], r1[2], r1[3]};
    v4f o2 = (v4f){r2[0], r2[1], r2[2], r2[3]};

    // Write-once output: non-temporal b128 stores (TH=NT), one per plane.
    size_t obase = (size_t)b * 3 * kHW + (size_t)y * kW + x0;
    __builtin_nontemporal_store(o0, (v4f*)(out + obase));
    __builtin_nontemporal_store(o1, (v4f*)(out + obase + kHW));
    __builtin_nontemporal_store(o2, (v4f*)(out + obase + 2 * (size_t)kHW));
}

extern "C" void kernel_launch(void* const* d_in, const int* in_sizes, int n_in,
                              void* d_out, int out_size, void* d_ws, size_t ws_size,
                              hipStream_t stream) {
    const float* depth = (const float*)d_in[0];  // (B,1,H,W) f32
    const float* intr  = (const float*)d_in[1];  // (B,3,3)   f32
    float*       out   = (float*)d_out;          // (B,3,H,W) f32

    dim3 grid(1, kH, kB);   // one block per image row
    dim3 block(kW4);        // 320 threads = 10 wave32
    SurfaceNormalOptimizer_kernel<<<grid, block, 0, stream>>>(depth, intr, out);
}